// SICLMMSEEqualizer_1271310320190
// MI455X (gfx1250) — compile-verified
//
#include <hip/hip_runtime.h>
#include <hip/hip_bf16.h>

// ---------------- problem dims (static, from reference) ----------------
#define B_    8
#define A_    8          // rx antennas
#define K_    4          // NTX*S streams
#define T_    14
#define F_    2048
#define NT    (B_*T_*F_)        // 229376 tones
#define TILE_F 64               // tones (f) per workgroup
#define NFB    (F_/TILE_F)      // 32 f-tiles per (b,t)
#define ROWS   (A_*K_)          // 32 h-rows per tile
#define HROWSTRIDE (T_*F_)      // 28672 complex elems between (a,k) rows
#define ROWDW  (TILE_F*2 + 2)   // 130 dwords: TDM pads 2 DW per 128-DW row
#define CH     16               // C-scratch chunk (tones)
#define CPAD   33               // padded C row (floats), odd -> conflict-free
#define INV_SQRT2f 0.70710678118654752f

typedef __attribute__((ext_vector_type(4))) unsigned int v4u_t;
typedef __attribute__((ext_vector_type(8))) int          v8i_t;
typedef __attribute__((ext_vector_type(4))) int          v4i_t;
typedef __attribute__((ext_vector_type(2))) float        v2f_t;
typedef __attribute__((ext_vector_type(8))) float        v8f_t;

// padded-LDS h fetch: complex at (row r = a*K+k, col)
__device__ __forceinline__ float2 ldh(const float* shf, int r, int col) {
    return make_float2(shf[r*ROWDW + col*2], shf[r*ROWDW + col*2 + 1]);
}

// =====================================================================
// Phase A (per SIC round):
//   - TDM DMAs the 32-row x TILE_F complex h tile into LDS (padded rows).
//   - Gram step G = H H^H + no*I via V_WMMA_F32_16X16X4_F32:
//       S = [Hr; Hi] (16x4), C = S*S^T -> ReG = C00+C11, ImG = C10-C01.
//     B operand == A operand register-for-register (S^T striping mirrors S).
//   - Per-lane complex Cholesky + 4-RHS triangular solves (pure VALU,
//     co-executes with the next wave's WMMA/LDS marshaling).
// =====================================================================
__global__ __launch_bounds__(TILE_F) void sic_phaseA(
    const float* __restrict__ hglob,      // h complex64 interleaved [B,1,A,K,T,F]
    const float* __restrict__ no_p,
    const float2* __restrict__ curr_y,    // [b][t][a][f]
    float2* __restrict__ z_st,            // [k][tone]
    float*  __restrict__ ne_st,           // [k][tone]
    float*  __restrict__ norms,           // [iter][K_]
    const int* __restrict__ sel, int iter)
{
    __shared__ float shf[ROWS * ROWDW];            // 16.64 KB padded h tile
    __shared__ float cbuf[2 * 8 * CH * CPAD];      // 33.8 KB C scratch (2 waves)

    const int tid = threadIdx.x;
    const int l   = tid & 31;       // lane
    const int wv  = tid >> 5;       // wave in block (0/1)
    const int fb  = blockIdx.x % NFB;
    const int bt  = blockIdx.x / NFB;
    const int t   = bt % T_;
    const int b   = bt / T_;
    const int f0  = fb * TILE_F;

#if defined(__HIP_DEVICE_COMPILE__)
    if (tid < 32) {   // wave 0: one TDM issue per workgroup
        unsigned long long gaddr = (unsigned long long)(uintptr_t)hglob +
            8ull * ((unsigned long long)b * (A_*K_*T_*F_) +
                    (unsigned long long)t * F_ + (unsigned)f0);
        unsigned lds_off = (unsigned)(unsigned long long)(uintptr_t)&shf[0];

        // D# group0: count=1, lds_addr, global_addr[56:0], type=2
        v4u_t g0 = { 1u, lds_off,
                     (unsigned)(gaddr & 0xffffffffu),
                     (unsigned)(((gaddr >> 32) & 0x01ffffffu) | 0x80000000u) };
        // D# group1: data_size=8B, pad_enable, pad_interval=128DW(code 6),
        //            pad_amount=2DW(code 1); tensor_dim0=tile_dim0=TILE_F,
        //            tensor_dim1=tile_dim1=32, dim0_stride=T*F
        v8i_t g1 = { (int)0x03930000u,
                     (int)((unsigned)TILE_F << 16),
                     (int)(32u << 16),
                     (int)((unsigned)TILE_F << 16),
                     (int)32,
                     (int)HROWSTRIDE,
                     0, 0 };
        v4i_t g2 = { 0, 0, 0, 0 };
        v4i_t g3 = { 0, 0, 0, 0 };
#if __has_include(<hip/amd_detail/amd_gfx1250_TDM.h>)
        v8i_t g4 = { 0, 0, 0, 0, 0, 0, 0, 0 };
        __builtin_amdgcn_tensor_load_to_lds(g0, g1, g2, g3, g4, 0);
#else
        __builtin_amdgcn_tensor_load_to_lds(g0, g1, g2, g3, 0);
#endif
        __builtin_amdgcn_s_wait_tensorcnt(0);
    }
#endif
    __syncthreads();

    // ---- active-stream mask ----
    bool act[K_];
    unsigned actm = 0;
#pragma unroll
    for (int k = 0; k < K_; ++k) act[k] = true;
    for (int j = 0; j < iter; ++j) {
        int s = sel[j];
#pragma unroll
        for (int k = 0; k < K_; ++k) if (k == s) act[k] = false;
    }
#pragma unroll
    for (int k = 0; k < K_; ++k) actm |= (act[k] ? 1u : 0u) << k;

    const float no = *no_p;
    float Lr[A_][A_], Li[A_][A_];

#if defined(__HIP_DEVICE_COMPILE__) && __has_builtin(__builtin_amdgcn_wmma_f32_16x16x4_f32)
    // ================= WMMA Gram build =================
    {
        float* cw = cbuf + wv * (8 * CH * CPAD);
        // A/B operand element for this lane: S[M][k], M = l&15,
        // k = vgpr + (l<16 ? 0 : 2); M<8 -> Re h[M][k], else Im h[M-8][k]
        const int M     = l & 15;
        const int kbase = (l < 16) ? 0 : 2;
        const int aRow  = (M < 8) ? M : (M - 8);
        const int riSel = (M < 8) ? 0 : 1;
        const bool on0  = (actm >> kbase) & 1u;
        const bool on1  = (actm >> (kbase + 1)) & 1u;
        const int ebase = (aRow*K_ + kbase)*ROWDW + riSel;

#pragma unroll
        for (int chunk = 0; chunk < 32/CH; ++chunk) {
#pragma unroll
            for (int tc = 0; tc < CH; ++tc) {
                const int fl = wv*32 + chunk*CH + tc;
                float e0 = shf[ebase + fl*2];
                float e1 = shf[ebase + ROWDW + fl*2];   // k+1 -> next h row
                e0 = on0 ? e0 : 0.f;
                e1 = on1 ? e1 : 0.f;
                v2f_t ab = { e0, e1 };
                v8f_t cz = { 0.f,0.f,0.f,0.f,0.f,0.f,0.f,0.f };
                v8f_t c  = __builtin_amdgcn_wmma_f32_16x16x4_f32(
                               false, ab, false, ab, (short)0, cz, false, false);
#pragma unroll
                for (int r = 0; r < 8; ++r)
                    cw[(r*CH + tc)*CPAD + l] = c[r];
            }
            asm volatile("" ::: "memory");   // intra-wave LDS is in-order in HW
            if ((l >> 4) == chunk) {         // this half-wave owns these tones
                const int tc = l & 15;
#pragma unroll
                for (int i = 0; i < A_; ++i) {
                    const int rb = (i*CH + tc)*CPAD;
#pragma unroll
                    for (int j = 0; j <= i; ++j) {
                        float c00 = cw[rb + j];          // Hr Hr^T
                        float c01 = cw[rb + 8  + j];     // Hr Hi^T
                        float c10 = cw[rb + 16 + j];     // Hi Hr^T
                        float c11 = cw[rb + 24 + j];     // Hi Hi^T
                        Lr[i][j] = c00 + c11 + ((i == j) ? no : 0.f);
                        Li[i][j] = c10 - c01;
                    }
                }
            }
            asm volatile("" ::: "memory");
        }
    }
#else
    // ================= VALU Gram build (fallback) =================
    {
        float hr[A_][K_], hi[A_][K_];
#pragma unroll
        for (int a = 0; a < A_; ++a)
#pragma unroll
            for (int k = 0; k < K_; ++k) {
                float2 v = ldh(shf, a*K_ + k, tid);
                hr[a][k] = act[k] ? v.x : 0.f;
                hi[a][k] = act[k] ? v.y : 0.f;
            }
#pragma unroll
        for (int i = 0; i < A_; ++i)
#pragma unroll
            for (int j = 0; j <= i; ++j) {
                float sr = (i == j) ? no : 0.f, si = 0.f;
#pragma unroll
                for (int k = 0; k < K_; ++k) {
                    sr += hr[i][k]*hr[j][k] + hi[i][k]*hi[j][k];
                    si += hi[i][k]*hr[j][k] - hr[i][k]*hi[j][k];
                }
                Lr[i][j] = sr; Li[i][j] = si;
            }
    }
#endif

    // ---- y for this tone ----
    const int fidx = f0 + tid;
    const int tone = bt * F_ + fidx;
    float yr[A_], yi[A_];
#pragma unroll
    for (int a = 0; a < A_; ++a) {
        float2 v = curr_y[(bt*A_ + a)*F_ + fidx];
        yr[a] = v.x; yi[a] = v.y;
    }

    // ---- complex Cholesky in place: G = L L^H ----
    float invd[A_];
#pragma unroll
    for (int j = 0; j < A_; ++j) {
        float d = Lr[j][j];
#pragma unroll
        for (int p = 0; p < j; ++p) d -= Lr[j][p]*Lr[j][p] + Li[j][p]*Li[j][p];
        d = fmaxf(d, 1e-30f);
        float r = rsqrtf(d);
        invd[j] = r;
        Lr[j][j] = d * r; Li[j][j] = 0.f;
#pragma unroll
        for (int i = j + 1; i < A_; ++i) {
            float sr = Lr[i][j], si = Li[i][j];
#pragma unroll
            for (int p = 0; p < j; ++p) {
                sr -= Lr[i][p]*Lr[j][p] + Li[i][p]*Li[j][p];
                si -= Li[i][p]*Lr[j][p] - Lr[i][p]*Li[j][p];
            }
            Lr[i][j] = sr * r; Li[i][j] = si * r;
        }
    }

    // ---- per stream k: solve G x = h_k (g_k = x^H); z, no_eff, |g|^2 ----
#pragma unroll
    for (int k = 0; k < K_; ++k) {
        const bool on = act[k];
        float wr[A_], wi[A_];
#pragma unroll
        for (int i = 0; i < A_; ++i) {            // forward: L w = h_k
            float2 v = ldh(shf, i*K_ + k, tid);
            float sr = on ? v.x : 0.f, si = on ? v.y : 0.f;
#pragma unroll
            for (int p = 0; p < i; ++p) {
                sr -= Lr[i][p]*wr[p] - Li[i][p]*wi[p];
                si -= Lr[i][p]*wi[p] + Li[i][p]*wr[p];
            }
            wr[i] = sr * invd[i]; wi[i] = si * invd[i];
        }
        float xr[A_], xi[A_];
#pragma unroll
        for (int i = A_ - 1; i >= 0; --i) {       // back: L^H x = w
            float sr = wr[i], si = wi[i];
#pragma unroll
            for (int p = i + 1; p < A_; ++p) {
                sr -= Lr[p][i]*xr[p] + Li[p][i]*xi[p];
                si -= Lr[p][i]*xi[p] - Li[p][i]*xr[p];
            }
            xr[i] = sr * invd[i]; xi[i] = si * invd[i];
        }
        float cnorm = 0.f, dr = 0.f, di = 0.f, zr = 0.f, zi = 0.f;
#pragma unroll
        for (int i = 0; i < A_; ++i) {
            float2 v = ldh(shf, i*K_ + k, tid);
            cnorm += xr[i]*xr[i] + xi[i]*xi[i];
            dr += v.x*xr[i] + v.y*xi[i];          // d = h_k^H x
            di += v.x*xi[i] - v.y*xr[i];
            zr += xr[i]*yr[i] + xi[i]*yi[i];      // z = conj(x) . y
            zi += xr[i]*yi[i] - xi[i]*yr[i];
        }
        float dd = dr*dr + di*di;
        float ne = on ? (dr / dd - 1.f) : 0.f;
        z_st[k*NT + tone] = make_float2(zr, zi);
        ne_st[k*NT + tone] = ne;

        // wave32 butterfly reduce of |g_k|^2, one atomic per wave
        float v = on ? cnorm : 0.f;
        v += __shfl_xor(v, 16, 32);
        v += __shfl_xor(v,  8, 32);
        v += __shfl_xor(v,  4, 32);
        v += __shfl_xor(v,  2, 32);
        v += __shfl_xor(v,  1, 32);
        if (l == 0) atomicAdd(&norms[iter*K_ + k], v);
    }
}

// =====================================================================
// Select: 1-thread argmin over active streams' accumulated norms.
// =====================================================================
__global__ void sic_select(const float* __restrict__ norms,
                           int* __restrict__ sel, int iter)
{
    if (threadIdx.x == 0 && blockIdx.x == 0) {
        float best = 3.4e38f; int bm = -1;
        for (int k = 0; k < K_; ++k) {
            bool used = false;
            for (int j = 0; j < iter; ++j) if (sel[j] == k) used = true;
            if (used) continue;
            float v = norms[iter*K_ + k];
            if (v < best) { best = v; bm = k; }
        }
        if (bm < 0) {
            for (int k = 0; k < K_ && bm < 0; ++k) {
                bool used = false;
                for (int j = 0; j < iter; ++j) if (sel[j] == k) used = true;
                if (!used) bm = k;
            }
        }
        sel[iter] = bm;
    }
}

// =====================================================================
// Phase B: QPSK hard slice of z_m, emit x/no_eff outputs, cancel h_m.
// =====================================================================
__global__ __launch_bounds__(256) void sic_phaseB(
    const float* __restrict__ hglob,
    const float2* __restrict__ z_st,
    const float*  __restrict__ ne_st,
    float2* __restrict__ curr_y,
    float*  __restrict__ out,
    const int* __restrict__ sel, int iter)
{
    int idx = blockIdx.x * 256 + threadIdx.x;
    if (idx >= NT) return;
    const int m  = sel[iter];
    const int f  = idx % F_;
    const int bt = idx / F_;
    const int t  = bt % T_;
    const int b  = bt / T_;

    float2 z = z_st[m*NT + idx];
    float  ne = ne_st[m*NT + idx];
    float xr = ((z.x > 0.f) ? 1.f : ((z.x < 0.f) ? -1.f : 0.f)) * INV_SQRT2f;
    float xi = ((z.y > 0.f) ? 1.f : ((z.y < 0.f) ? -1.f : 0.f)) * INV_SQRT2f;

    const int o = ((b*K_ + m)*T_ + t)*F_ + f;
    out[2*o]     = xr;
    out[2*o + 1] = xi;
    out[2*(B_*K_*T_*F_) + o] = ne;

#pragma unroll
    for (int a = 0; a < A_; ++a) {
        int hoff = (((b*A_ + a)*K_ + m)*T_ + t)*F_ + f;
        float2 hv = ((const float2*)hglob)[hoff];
        int cyo = (bt*A_ + a)*F_ + f;
        float2 cv = curr_y[cyo];
        cv.x -= hv.x*xr - hv.y*xi;
        cv.y -= hv.x*xi + hv.y*xr;
        curr_y[cyo] = cv;
    }
}

// =====================================================================
// Init: transpose y[B,1,A,T,F] -> curr_y[b][t][a][f]; zero norms/sel.
// =====================================================================
__global__ void sic_init(const float2* __restrict__ y,
                         float2* __restrict__ curr_y,
                         float* __restrict__ norms, int* __restrict__ sel)
{
    int idx = blockIdx.x * 256 + threadIdx.x;
    if (idx < NT * A_) {
        int f = idx % F_;  int r = idx / F_;
        int a = r % A_;    int r2 = r / A_;
        int t = r2 % T_;   int b = r2 / T_;
        curr_y[((b*T_ + t)*A_ + a)*F_ + f] = y[((b*A_ + a)*T_ + t)*F_ + f];
    }
    if (blockIdx.x == 0 && threadIdx.x < K_*K_) norms[threadIdx.x] = 0.f;
    if (blockIdx.x == 0 && threadIdx.x < K_)    sel[threadIdx.x]   = -1;
}

// =====================================================================
extern "C" void kernel_launch(void* const* d_in, const int* in_sizes, int n_in,
                              void* d_out, int out_size, void* d_ws, size_t ws_size,
                              hipStream_t stream)
{
    (void)in_sizes; (void)n_in; (void)out_size; (void)ws_size;
    const float2* y  = (const float2*)d_in[0];
    const float*  h  = (const float*)d_in[1];
    const float*  no = (const float*)d_in[3];   // err_var (d_in[2]) unused by math
    float* out = (float*)d_out;

    char* w = (char*)d_ws;                      // ~25.8 MB workspace
    float2* curr_y = (float2*)w;  w += (size_t)NT * A_ * sizeof(float2);
    float2* z_st   = (float2*)w;  w += (size_t)K_ * NT * sizeof(float2);
    float*  ne_st  = (float*)w;   w += (size_t)K_ * NT * sizeof(float);
    float*  norms  = (float*)w;   w += 64;
    int*    sel    = (int*)w;

    const int initBlocks = (NT * A_ + 255) / 256;
    const int blocksA = B_ * T_ * NFB;          // 3584
    const int blocksB = (NT + 255) / 256;

    sic_init<<<initBlocks, 256, 0, stream>>>(y, curr_y, norms, sel);
    for (int it = 0; it < K_; ++it) {
        sic_phaseA<<<blocksA, TILE_F, 0, stream>>>(h, no, curr_y, z_st, ne_st,
                                                   norms, sel, it);
        sic_select<<<1, 32, 0, stream>>>(norms, sel, it);
        sic_phaseB<<<blocksB, 256, 0, stream>>>(h, z_st, ne_st, curr_y, out,
                                                sel, it);
    }
}